// SheafGCNLayer3_79027398246777
// MI455X (gfx1250) — compile-verified
//
#include <hip/hip_runtime.h>

// SheafGCN / RGCN layer for MI455X (gfx1250, wave32).
//
//   x_all[n,t,:] = x[n,:] @ weight[t]          (8x per-type GEMM, WMMA f32 16x16x4)
//   out[n,:]     = x[n,:] @ self_loop_w^T      (self-loop GEMM, WMMA f32 16x16x4)
//   out[dst,:]  += x_all[src,etype,:]          (per-edge gather + f32 atomic scatter)
//
// Memory-bound problem (~0.8 ms of traffic at 23.3 TB/s); x_all (205 MB) is
// nearly L2-resident (192 MB), so edge gathers mostly hit L2.

#define D_IN     128
#define D_OUT    128
#define N_TYPES  8
#define XS_STRIDE 132   // padded LDS row stride: 132%64=4 -> conflict-free, 16B aligned

typedef float v2f __attribute__((ext_vector_type(2)));
typedef float v8f __attribute__((ext_vector_type(8)));

// ---------------------------------------------------------------------------
// Per-wave 16x16 f32 WMMA tile: C += A(16x4) * B(4x16) over K=128 in steps of 4.
// A-fragment (ISA 7.12.2, 32-bit A 16x4): lane half h, VGPR j -> K = k0+2h+j,
// row M = lane&15  => one contiguous float2 per lane from the LDS x-tile.
// B-fragment (mirror layout): lane half h, VGPR j -> K = k0+2h+j, col N = lane&15.
// ---------------------------------------------------------------------------

__global__ __launch_bounds__(288) void rgcn_gemm_kernel(
    const float* __restrict__ x,      // [N, 128]
    const float* __restrict__ w,      // [8, 128, 128]
    const float* __restrict__ slw,    // [128, 128]  (torch Linear layout: [d_out, d_in])
    float*       __restrict__ x_all,  // [N, 8, 128] workspace
    float*       __restrict__ out)    // [N, 128]
{
    __shared__ float xs[16 * XS_STRIDE];

    const int node_base = blockIdx.x * 16;
    const int tid = threadIdx.x;

    // Cooperative load of the 16x128 x tile (waves 0..7; 2 float4 per thread).
    if (tid < 256) {
        const int row = tid >> 4;
        const int col = (tid & 15) * 8;
        const float4* src = (const float4*)(x + (size_t)(node_base + row) * D_IN + col);
        float4 lo = src[0];
        float4 hi = src[1];
        *(float4*)&xs[row * XS_STRIDE + col]     = lo;
        *(float4*)&xs[row * XS_STRIDE + col + 4] = hi;
    }
    __syncthreads();

    const int wave = tid >> 5;          // 0..7: edge-type GEMMs, 8: self-loop GEMM
    const int lane = tid & 31;
    const int half = lane >> 4;         // lane half selects K sub-block
    const int l16  = lane & 15;

    if (wave < N_TYPES) {
        const float* wt = w + (size_t)wave * D_IN * D_OUT;   // [128 x 128], row-major [k][n]
        for (int ct = 0; ct < D_OUT / 16; ++ct) {
            const int colb = ct * 16;
            v8f acc = {};
            #pragma unroll 4
            for (int k0 = 0; k0 < D_IN; k0 += 4) {
                const int ka = k0 + 2 * half;
                v2f a = *(const v2f*)&xs[l16 * XS_STRIDE + ka];
                v2f b;
                b.x = wt[(size_t)(ka + 0) * D_OUT + colb + l16];
                b.y = wt[(size_t)(ka + 1) * D_OUT + colb + l16];
                acc = __builtin_amdgcn_wmma_f32_16x16x4_f32(
                    false, a, false, b, (short)0, acc, false, false);
            }
            // C/D layout: VGPR v -> row v + 8*half, col l16
            #pragma unroll
            for (int v = 0; v < 8; ++v) {
                const int m = v + 8 * half;
                x_all[(((size_t)(node_base + m) * N_TYPES + wave) << 7) + colb + l16] = acc[v];
            }
        }
    } else {
        // Self-loop: out_tile = x_tile @ slw^T   => B[k][n] = slw[n][k] (contiguous in k)
        for (int ct = 0; ct < D_OUT / 16; ++ct) {
            const int colb = ct * 16;
            v8f acc = {};
            #pragma unroll 4
            for (int k0 = 0; k0 < D_IN; k0 += 4) {
                const int ka = k0 + 2 * half;
                v2f a = *(const v2f*)&xs[l16 * XS_STRIDE + ka];
                v2f b = *(const v2f*)&slw[(size_t)(colb + l16) * D_IN + ka];
                acc = __builtin_amdgcn_wmma_f32_16x16x4_f32(
                    false, a, false, b, (short)0, acc, false, false);
            }
            #pragma unroll
            for (int v = 0; v < 8; ++v) {
                const int m = v + 8 * half;
                out[((size_t)(node_base + m) << 7) + colb + l16] = acc[v];
            }
        }
    }
}

// ---------------------------------------------------------------------------
// Edge scatter: one edge per wave32. All 32 lanes load the 512B message row
// (float4, fully coalesced) and atomically add into out[dst].
// ---------------------------------------------------------------------------
__global__ __launch_bounds__(256) void rgcn_edge_scatter_kernel(
    const float* __restrict__ x_all,  // [N, 8, 128]
    const int*   __restrict__ ei,     // [2, E] (src row then dst row)
    const int*   __restrict__ et,     // [E]
    float*       __restrict__ out,    // [N, 128]
    int n_edges)
{
    const int wave = blockIdx.x * (blockDim.x >> 5) + (threadIdx.x >> 5);
    const int lane = threadIdx.x & 31;
    if (wave >= n_edges) return;

    const int src = ei[wave];
    const int dst = ei[n_edges + wave];
    const int t   = et[wave];

    const float4* msg = (const float4*)(x_all + (((size_t)src * N_TYPES + t) << 7));
    float4 v = msg[lane];

    float* o = out + (((size_t)dst) << 7) + lane * 4;
    atomicAdd(o + 0, v.x);
    atomicAdd(o + 1, v.y);
    atomicAdd(o + 2, v.z);
    atomicAdd(o + 3, v.w);
}

extern "C" void kernel_launch(void* const* d_in, const int* in_sizes, int n_in,
                              void* d_out, int out_size, void* d_ws, size_t ws_size,
                              hipStream_t stream) {
    const float* x   = (const float*)d_in[0];   // [N, 128]
    const float* w   = (const float*)d_in[1];   // [8, 128, 128]
    const float* slw = (const float*)d_in[2];   // [128, 128]
    const int*   ei  = (const int*)d_in[3];     // [2, E]
    const int*   et  = (const int*)d_in[4];     // [E]
    float* out = (float*)d_out;
    float* x_all = (float*)d_ws;                // N*8*128 floats (~205 MB)

    const int n_nodes = in_sizes[0] / D_IN;     // 50000 (multiple of 16)
    const int n_edges = in_sizes[4];            // 600000

    // GEMMs: one block per 16-node tile, 9 waves (8 type GEMMs + self-loop).
    rgcn_gemm_kernel<<<n_nodes / 16, 288, 0, stream>>>(x, w, slw, x_all, out);

    // Edge gather + atomic scatter (ordered after GEMM by stream).
    const int waves_per_block = 256 / 32;
    const int blocks = (n_edges + waves_per_block - 1) / waves_per_block;
    rgcn_edge_scatter_kernel<<<blocks, 256, 0, stream>>>(x_all, ei, et, out, n_edges);
}